// GraphRNA_58772332478774
// MI455X (gfx1250) — compile-verified
//
#include <hip/hip_runtime.h>

typedef float v2f __attribute__((ext_vector_type(2)));
typedef float v8f __attribute__((ext_vector_type(8)));

#define H 128
#define TPAD 132   // LDS row stride (128 + 4) to break bank conflicts
#define RT 4       // row tiles (of 16) per GEMM block -> 64 rows/block

// ---------------------------------------------------------------------------
// degree counting: cnt[idx[e]] += 1
__global__ void count_edges_k(const int* __restrict__ idx, int E, float* __restrict__ cnt) {
    int e = blockIdx.x * blockDim.x + threadIdx.x;
    if (e < E) atomicAdd(&cnt[idx[e]], 1.0f);
}

// v <- 1 / max(v, 1)   (SAGE mean divisor)
__global__ void recip_max1_k(float* __restrict__ v, int n) {
    int i = blockIdx.x * blockDim.x + threadIdx.x;
    if (i < n) v[i] = 1.0f / fmaxf(v[i], 1.0f);
}

// v <- rsqrt(v + 1)    (GCN deg^-1/2 including self loop; deg+1 >= 1 always)
__global__ void dinv_k(float* __restrict__ v, int n) {
    int i = blockIdx.x * blockDim.x + threadIdx.x;
    if (i < n) v[i] = rsqrtf(v[i] + 1.0f);
}

// out[r, j] = b0[j] (+ b1[j] + b2[j] + b3[j])  -- bias broadcast init
__global__ void init_bias_k(float* __restrict__ out, int n,
                            const float* __restrict__ b0, const float* __restrict__ b1,
                            const float* __restrict__ b2, const float* __restrict__ b3) {
    long long i = (long long)blockIdx.x * blockDim.x + threadIdx.x;
    if (i < (long long)n * H) {
        int j = (int)(i & (H - 1));
        float v = b0[j];
        if (b1) v += b1[j];
        if (b2) v += b2[j];
        if (b3) v += b3[j];
        out[i] = v;
    }
}

// agg[dst[e], :] += x[src[e], :]
// one wave32 per edge iteration, float4 per lane; grid-stride with prefetch of the
// next iteration's source row (global_prefetch_b8) to hide random-gather latency
// behind the L2 atomic stream.
__global__ void scatter_rows_k(const float* __restrict__ x, const int* __restrict__ src,
                               const int* __restrict__ dst, int E, float* __restrict__ agg) {
    int lane = threadIdx.x & 31;
    int wid  = (blockIdx.x * blockDim.x + threadIdx.x) >> 5;
    int nw   = (gridDim.x * blockDim.x) >> 5;
    for (int e = wid; e < E; e += nw) {
        int s = src[e];
        int d = dst[e];
        int en = e + nw;
        if (en < E) {
            int sn = src[en];
            __builtin_prefetch(x + (size_t)sn * H + lane * 4, 0, 3);
        }
        float4 v = ((const float4*)(x + (size_t)s * H))[lane];
        float* o = agg + (size_t)d * H + lane * 4;
        atomicAdd(o + 0, v.x);
        atomicAdd(o + 1, v.y);
        atomicAdd(o + 2, v.z);
        atomicAdd(o + 3, v.w);
    }
}

// OUT[n,128] += rowscale[r] * (X[n,128] @ W[128,128]); optional fused ReLU on the
// final accumulated value. Block: 256 threads = 8 waves; block computes 64 rows
// (4 row-tiles), wave w owns cols [16w,16w+16). Each B fragment feeds 4 WMMAs.
__global__ void __launch_bounds__(256)
gemm128_k(const float* __restrict__ X, const float* __restrict__ W,
          float* __restrict__ OUT, int n, const float* __restrict__ rowscale,
          int do_relu) {
    __shared__ float tile[RT * 16 * TPAD];
    int rowBase = blockIdx.x * (RT * 16);
    for (int i = threadIdx.x; i < RT * 16 * H; i += 256) {
        int r = i >> 7, c = i & 127;
        tile[r * TPAD + c] = (rowBase + r < n) ? X[(size_t)(rowBase + r) * H + c] : 0.0f;
    }
    __syncthreads();

    int wave = threadIdx.x >> 5;
    int lane = threadIdx.x & 31;
    int half = lane >> 4;    // 0: K%4 in {0,1}; 1: K%4 in {2,3}
    int lid  = lane & 15;
    int colBase = wave * 16;

    v8f acc[RT];
#pragma unroll
    for (int t = 0; t < RT; t++) acc[t] = (v8f){};

#pragma unroll
    for (int k = 0; k < H; k += 4) {
        int ka = k + 2 * half;
        // B 4x16 f32 fragment: rows striped across lanes, mirrored halves
        v2f b;
        b.x = W[(size_t)ka * H + colBase + lid];
        b.y = W[(size_t)(ka + 1) * H + colBase + lid];
#pragma unroll
        for (int t = 0; t < RT; t++) {
            // A 16x4 f32 fragment: lanes0-15 -> {K=k,k+1}; lanes16-31 -> {K=k+2,k+3}
            v2f a;
            a.x = tile[(t * 16 + lid) * TPAD + ka];
            a.y = tile[(t * 16 + lid) * TPAD + ka + 1];
            acc[t] = __builtin_amdgcn_wmma_f32_16x16x4_f32(false, a, false, b,
                                                           (short)0, acc[t], false, false);
        }
    }
    // C/D layout: lanes0-15 VGPRv -> M=v; lanes16-31 -> M=v+8; N=lid
#pragma unroll
    for (int t = 0; t < RT; t++) {
#pragma unroll
        for (int v = 0; v < 8; v++) {
            int r = rowBase + t * 16 + v + 8 * half;
            if (r < n) {
                float val = acc[t][v];
                if (rowscale) val *= rowscale[r];
                float* p = OUT + (size_t)r * H + colBase + lid;
                float res = *p + val;
                if (do_relu) res = fmaxf(res, 0.0f);
                *p = res;
            }
        }
    }
}

// out[r,:] += scale[r] * g[r,:], optional fused ReLU
__global__ void axpy_rowscale_k(float* __restrict__ out, const float* __restrict__ g,
                                const float* __restrict__ scale, int n, int do_relu) {
    long long i = (long long)blockIdx.x * blockDim.x + threadIdx.x;
    if (i < (long long)n * H) {
        float res = out[i] + scale[i >> 7] * g[i];
        if (do_relu) res = fmaxf(res, 0.0f);
        out[i] = res;
    }
}

__global__ void addW_k(const float* __restrict__ a, const float* __restrict__ b,
                       float* __restrict__ o) {
    int i = blockIdx.x * blockDim.x + threadIdx.x;
    if (i < H * H) o[i] = a[i] + b[i];
}

// out[p] = dot(A[ia[p],:], B[ib[p],:])  -- one wave32 per pair
__global__ void dot_pairs_k(const float* __restrict__ A, const int* __restrict__ ia,
                            const float* __restrict__ B, const int* __restrict__ ib,
                            float* __restrict__ out, int L) {
    int p = (blockIdx.x * blockDim.x + threadIdx.x) >> 5;
    int lane = threadIdx.x & 31;
    if (p >= L) return;
    float4 a = ((const float4*)(A + (size_t)ia[p] * H))[lane];
    float4 b = ((const float4*)(B + (size_t)ib[p] * H))[lane];
    float s = a.x * b.x + a.y * b.y + a.z * b.z + a.w * b.w;
#pragma unroll
    for (int off = 16; off > 0; off >>= 1) s += __shfl_xor(s, off, 32);
    if (lane == 0) out[p] = s;
}

// ---------------------------------------------------------------------------
static inline int scatter_blocks(int E) {
    int b = (E + 7) / 8;           // 8 waves per 256-thread block
    return b > 4096 ? 4096 : b;    // cap -> grid-stride loop + prefetch kicks in
}

extern "C" void kernel_launch(void* const* d_in, const int* in_sizes, int n_in,
                              void* d_out, int out_size, void* d_ws, size_t ws_size,
                              hipStream_t stream) {
    const float* emb_s   = (const float*)d_in[0];
    const float* emb_m   = (const float*)d_in[1];
    const float* emb_r   = (const float*)d_in[2];
    const float* sage_Wl = (const float*)d_in[3];
    const float* sage_bl = (const float*)d_in[4];
    const float* sage_Wr = (const float*)d_in[5];
    const float* gcn_Wp  = (const float*)d_in[6];
    const float* gcn_bp  = (const float*)d_in[7];
    const int* src_sm  = (const int*)d_in[8];
    const int* dst_sm  = (const int*)d_in[9];
    const int* src_rm  = (const int*)d_in[10];
    const int* dst_rm  = (const int*)d_in[11];
    const int* src_sim = (const int*)d_in[12];
    const int* dst_sim = (const int*)d_in[13];
    const int* lbl_s   = (const int*)d_in[14];
    const int* lbl_m   = (const int*)d_in[15];
    const int* lbl_r   = (const int*)d_in[16];

    const int NS = in_sizes[0] / H, NM = in_sizes[1] / H, NR = in_sizes[2] / H;
    const int E_SM = in_sizes[8], E_RM = in_sizes[10], E_SIM = in_sizes[12];
    const int L = in_sizes[14];

    float* ws = (float*)d_ws;
    size_t off = 0;
    auto alloc = [&](size_t nelem) {
        float* p = ws + off;
        off += (nelem + 127) & ~(size_t)127;  // keep 512B row alignment for float4
        return p;
    };
    float* xs_a = alloc((size_t)NS * H);
    float* xm_a = alloc((size_t)NM * H);
    float* xr_a = alloc((size_t)NR * H);
    float* xs_b = alloc((size_t)NS * H);
    float* xm_b = alloc((size_t)NM * H);
    float* xr_b = alloc((size_t)NR * H);
    float* aggm = alloc((size_t)NM * H);
    float* zbuf = alloc((size_t)NM * H);
    float* gbuf = alloc((size_t)NM * H);
    float* aggs = alloc((size_t)NS * H);
    float* aggr = alloc((size_t)NR * H);
    float* Wsum = alloc((size_t)H * H);
    float* rs_m_sm = alloc(NM);
    float* rs_m_rm = alloc(NM);
    float* dinv0   = alloc(NM);
    float* dinv1   = alloc(NM);
    float* rs_s    = alloc(NS);
    float* rs_r    = alloc(NR);
    (void)ws_size; (void)n_in; (void)out_size;

    const long long nmH = (long long)NM * H;
    const long long nsH = (long long)NS * H;
    const long long nrH = (long long)NR * H;
    const int gbM = (int)((nmH + 255) / 256);
    const int gbS = (int)((nsH + 255) / 256);
    const int gbR = (int)((nrH + 255) / 256);
    const int gmM = (NM + RT * 16 - 1) / (RT * 16);
    const int gmS = (NS + RT * 16 - 1) / (RT * 16);
    const int gmR = (NR + RT * 16 - 1) / (RT * 16);

    // --- edge-structure precompute (layer independent) ---
    hipMemsetAsync(rs_m_sm, 0, (size_t)NM * 4, stream);
    hipMemsetAsync(rs_m_rm, 0, (size_t)NM * 4, stream);
    hipMemsetAsync(dinv0,   0, (size_t)NM * 4, stream);
    hipMemsetAsync(dinv1,   0, (size_t)NM * 4, stream);
    hipMemsetAsync(rs_s,    0, (size_t)NS * 4, stream);
    hipMemsetAsync(rs_r,    0, (size_t)NR * 4, stream);
    count_edges_k<<<(E_SM + 255) / 256, 256, 0, stream>>>(dst_sm, E_SM, rs_m_sm);
    count_edges_k<<<(E_RM + 255) / 256, 256, 0, stream>>>(dst_rm, E_RM, rs_m_rm);
    count_edges_k<<<(E_SM + 255) / 256, 256, 0, stream>>>(src_sm, E_SM, rs_s);
    count_edges_k<<<(E_RM + 255) / 256, 256, 0, stream>>>(src_rm, E_RM, rs_r);
    count_edges_k<<<(E_SIM + 255) / 256, 256, 0, stream>>>(dst_sim, E_SIM, dinv0);
    count_edges_k<<<(E_SIM + 255) / 256, 256, 0, stream>>>(src_sim, E_SIM, dinv1);
    recip_max1_k<<<(NM + 255) / 256, 256, 0, stream>>>(rs_m_sm, NM);
    recip_max1_k<<<(NM + 255) / 256, 256, 0, stream>>>(rs_m_rm, NM);
    recip_max1_k<<<(NS + 255) / 256, 256, 0, stream>>>(rs_s, NS);
    recip_max1_k<<<(NR + 255) / 256, 256, 0, stream>>>(rs_r, NR);
    dinv_k<<<(NM + 255) / 256, 256, 0, stream>>>(dinv0, NM);
    dinv_k<<<(NM + 255) / 256, 256, 0, stream>>>(dinv1, NM);

    const float *xs = emb_s, *xm = emb_m, *xr = emb_r;
    float *nxs = xs_a, *nxm = xm_a, *nxr = xr_a;

    for (int l = 0; l < 2; l++) {
        const float* Wl = sage_Wl + (size_t)l * 4 * H * H;
        const float* bl = sage_bl + (size_t)l * 4 * H;
        const float* Wr = sage_Wr + (size_t)l * 4 * H * H;
        const float* gW = gcn_Wp  + (size_t)l * 2 * H * H;
        const float* gb = gcn_bp  + (size_t)l * 2 * H;

        // ---- mRNA target: 2 SAGE + 2 GCN, all biases folded into init ----
        init_bias_k<<<gbM, 256, 0, stream>>>(nxm, NM, bl + 0 * H, bl + 2 * H,
                                             gb + 0 * H, gb + 1 * H);
        // SAGE srna->mrna
        hipMemsetAsync(aggm, 0, nmH * 4, stream);
        scatter_rows_k<<<scatter_blocks(E_SM), 256, 0, stream>>>(xs, src_sm, dst_sm, E_SM, aggm);
        gemm128_k<<<gmM, 256, 0, stream>>>(aggm, Wl + 0 * H * H, nxm, NM, rs_m_sm, 0);
        // SAGE rbp->mrna
        hipMemsetAsync(aggm, 0, nmH * 4, stream);
        scatter_rows_k<<<scatter_blocks(E_RM), 256, 0, stream>>>(xr, src_rm, dst_rm, E_RM, aggm);
        gemm128_k<<<gmM, 256, 0, stream>>>(aggm, Wl + 2 * H * H, nxm, NM, rs_m_rm, 0);
        // root transforms x_m @ (Wr0 + Wr2) fused into one GEMM
        addW_k<<<(H * H + 255) / 256, 256, 0, stream>>>(Wr + 0 * H * H, Wr + 2 * H * H, Wsum);
        gemm128_k<<<gmM, 256, 0, stream>>>(xm, Wsum, nxm, NM, nullptr, 0);
        // GCN forward: z = diag(dinv0)(x W0); g = z(selfloop) + scatter(z); out += diag(dinv0) g
        hipMemsetAsync(zbuf, 0, nmH * 4, stream);
        gemm128_k<<<gmM, 256, 0, stream>>>(xm, gW + 0 * H * H, zbuf, NM, dinv0, 0);
        hipMemcpyAsync(gbuf, zbuf, nmH * 4, hipMemcpyDeviceToDevice, stream);
        scatter_rows_k<<<scatter_blocks(E_SIM), 256, 0, stream>>>(zbuf, src_sim, dst_sim, E_SIM, gbuf);
        axpy_rowscale_k<<<gbM, 256, 0, stream>>>(nxm, gbuf, dinv0, NM, 0);
        // GCN reverse (edges dst_sim -> src_sim, degrees over src_sim); fused ReLU
        hipMemsetAsync(zbuf, 0, nmH * 4, stream);
        gemm128_k<<<gmM, 256, 0, stream>>>(xm, gW + 1 * H * H, zbuf, NM, dinv1, 0);
        hipMemcpyAsync(gbuf, zbuf, nmH * 4, hipMemcpyDeviceToDevice, stream);
        scatter_rows_k<<<scatter_blocks(E_SIM), 256, 0, stream>>>(zbuf, dst_sim, src_sim, E_SIM, gbuf);
        axpy_rowscale_k<<<gbM, 256, 0, stream>>>(nxm, gbuf, dinv1, NM, 1);

        // ---- sRNA target: SAGE mrna->srna over reversed sm edges; ReLU in last GEMM ----
        init_bias_k<<<gbS, 256, 0, stream>>>(nxs, NS, bl + 1 * H, nullptr, nullptr, nullptr);
        hipMemsetAsync(aggs, 0, nsH * 4, stream);
        scatter_rows_k<<<scatter_blocks(E_SM), 256, 0, stream>>>(xm, dst_sm, src_sm, E_SM, aggs);
        gemm128_k<<<gmS, 256, 0, stream>>>(aggs, Wl + 1 * H * H, nxs, NS, rs_s, 0);
        gemm128_k<<<gmS, 256, 0, stream>>>(xs, Wr + 1 * H * H, nxs, NS, nullptr, 1);

        // ---- RBP target: SAGE mrna->rbp over reversed rm edges; ReLU in last GEMM ----
        init_bias_k<<<gbR, 256, 0, stream>>>(nxr, NR, bl + 3 * H, nullptr, nullptr, nullptr);
        hipMemsetAsync(aggr, 0, nrH * 4, stream);
        scatter_rows_k<<<scatter_blocks(E_RM), 256, 0, stream>>>(xm, dst_rm, src_rm, E_RM, aggr);
        gemm128_k<<<gmR, 256, 0, stream>>>(aggr, Wl + 3 * H * H, nxr, NR, rs_r, 0);
        gemm128_k<<<gmR, 256, 0, stream>>>(xr, Wr + 3 * H * H, nxr, NR, nullptr, 1);

        // ping-pong
        xs = nxs; xm = nxm; xr = nxr;
        nxs = (l == 0) ? xs_b : xs_a;
        nxm = (l == 0) ? xm_b : xm_a;
        nxr = (l == 0) ? xr_b : xr_a;
    }

    // ---- link predictions ----
    float* out = (float*)d_out;
    dot_pairs_k<<<(L + 7) / 8, 256, 0, stream>>>(xs, lbl_s, xm, lbl_m, out, L);
    dot_pairs_k<<<(L + 7) / 8, 256, 0, stream>>>(xr, lbl_r, xm, lbl_m, out + L, L);
}